// MLFF_36773509989089
// MI455X (gfx1250) — compile-verified
//
#include <hip/hip_runtime.h>
#include <hip/hip_bf16.h>

// MLFF fused MLP (WMMA fp32) + gather-contract force kernel for gfx1250.
//
// Roofline: force pass streams dfeat (206 MB) once -> ~8.9us @ 23.3 TB/s; it is
// pure bandwidth. MLP fwd+bwd (~0.2 GFLOP) runs on the fp32 matrix pipe via
// V_WMMA_F32_16X16X4_F32 to keep full fp32 fidelity with the reference.

typedef __attribute__((ext_vector_type(2))) float v2f;
typedef __attribute__((ext_vector_type(8))) float v8f;

#define B_  4
#define N_  1024
#define K_  100
#define F_  42
#define H_  64
#define FP_ 44   // F padded to a multiple of 4 for the WMMA K-loop

// ---------------------------------------------------------------------------
// Kernel 1: fused MLP forward + backward for a 16-atom tile.
// 128 threads = 4 waves; wave w owns output columns [16w, 16w+16).
// WMMA fragment layouts (CDNA5 ISA 7.12.2, 32-bit, wave32):
//   A (16xK tile, per 16x16x4 step): M = lane&15, K = v + 2*(lane>>4)
//   B (Kx16):                        N = lane&15, K = v + 2*(lane>>4)
//   C/D (16x16):                     N = lane&15, M = v + 8*(lane>>4)
// ---------------------------------------------------------------------------
__global__ __launch_bounds__(128) void mlff_mlp_wmma_kernel(
    const float* __restrict__ image,   // [B*N, F]
    const float* __restrict__ W1,      // [F, H]
    const float* __restrict__ b1,      // [H]
    const float* __restrict__ W2,      // [H, H]
    const float* __restrict__ b2,      // [H]
    const float* __restrict__ W3,      // [H, 1]
    const float* __restrict__ b3,      // [1]
    float* __restrict__ Ei_out,        // [B*N]
    float* __restrict__ dE_out)        // [B*N, F]
{
  __shared__ float W1s[FP_][H_];   // K-padded rows 42..43 zeroed
  __shared__ float W2s[H_][H_];
  __shared__ float W3s[H_];
  __shared__ float Xs[16][FP_];    // cols 42..43 zeroed
  __shared__ float H1s[16][H_];
  __shared__ float H2s[16][H_];
  __shared__ float G2s[16][H_];
  __shared__ float G1s[16][H_];

  const int tid   = threadIdx.x;
  const int wave  = tid >> 5;
  const int lane  = tid & 31;
  const int half  = lane >> 4;
  const int l16   = lane & 15;
  const int col   = wave * 16 + l16;     // column this lane owns in the 64-wide slab
  const int atom0 = blockIdx.x * 16;     // flat (b*N + n) base

  // ---- stage weights + X tile in LDS ----
  for (int i = tid; i < FP_ * H_; i += 128) {
    int r = i / H_, c = i % H_;
    W1s[r][c] = (r < F_) ? W1[r * H_ + c] : 0.f;
  }
  for (int i = tid; i < H_ * H_; i += 128)
    W2s[i / H_][i % H_] = W2[i];
  if (tid < H_) W3s[tid] = W3[tid];
  for (int i = tid; i < 16 * FP_; i += 128) {
    int r = i / FP_, c = i % FP_;
    Xs[r][c] = (c < F_) ? image[(size_t)(atom0 + r) * F_ + c] : 0.f;
  }
  __syncthreads();

  // ---- GEMM1: H1pre = X @ W1  (16x42 @ 42x64, K padded to 44) ----
  v8f acc = {};
  for (int kb = 0; kb < FP_; kb += 4) {
    const int kk = kb + 2 * half;
    v2f a, b;
    a.x = Xs[l16][kk];      a.y = Xs[l16][kk + 1];
    b.x = W1s[kk][col];     b.y = W1s[kk + 1][col];
    acc = __builtin_amdgcn_wmma_f32_16x16x4_f32(false, a, false, b,
                                                (short)0, acc, false, false);
  }
  {
    const float bias = b1[col];
#pragma unroll
    for (int v = 0; v < 8; ++v)
      H1s[v + 8 * half][col] = tanhf(acc[v] + bias);
  }
  __syncthreads();

  // ---- GEMM2: H2pre = H1 @ W2  (16x64 @ 64x64) ----
  acc = (v8f){};
  for (int kb = 0; kb < H_; kb += 4) {
    const int kk = kb + 2 * half;
    v2f a, b;
    a.x = H1s[l16][kk];     a.y = H1s[l16][kk + 1];
    b.x = W2s[kk][col];     b.y = W2s[kk + 1][col];
    acc = __builtin_amdgcn_wmma_f32_16x16x4_f32(false, a, false, b,
                                                (short)0, acc, false, false);
  }
  {
    const float bias = b2[col];
#pragma unroll
    for (int v = 0; v < 8; ++v)
      H2s[v + 8 * half][col] = tanhf(acc[v] + bias);
  }
  __syncthreads();

  // ---- g2 = (1 - h2^2) * W3 ;  Ei = h2 . W3 + b3 ----
  for (int i = tid; i < 16 * H_; i += 128) {
    int r = i / H_, c = i % H_;
    float h = H2s[r][c];
    G2s[r][c] = (1.f - h * h) * W3s[c];
  }
  __syncthreads();
  if (tid < 16) {
    float s = b3[0];
    for (int j = 0; j < H_; ++j) s += H2s[tid][j] * W3s[j];
    Ei_out[atom0 + tid] = s;
  }

  // ---- GEMM3: g1pre = g2 @ W2^T  (B fragment read transposed from LDS) ----
  acc = (v8f){};
  for (int kb = 0; kb < H_; kb += 4) {
    const int kk = kb + 2 * half;
    v2f a, b;
    a.x = G2s[l16][kk];     a.y = G2s[l16][kk + 1];
    b.x = W2s[col][kk];     b.y = W2s[col][kk + 1];   // W2T[kk][col]
    acc = __builtin_amdgcn_wmma_f32_16x16x4_f32(false, a, false, b,
                                                (short)0, acc, false, false);
  }
#pragma unroll
  for (int v = 0; v < 8; ++v) {
    const int m = v + 8 * half;
    float h = H1s[m][col];
    G1s[m][col] = (1.f - h * h) * acc[v];
  }
  __syncthreads();

  // ---- GEMM4: dE = g1 @ W1^T  (16x64 @ 64x42, N padded to 48 -> waves 0..2) ----
  if (wave < 3) {
    acc = (v8f){};
    const int n = wave * 16 + l16;       // output feature column (0..47)
    const bool valid = (n < F_);
    for (int kb = 0; kb < H_; kb += 4) {
      const int kk = kb + 2 * half;
      v2f a, b;
      a.x = G1s[l16][kk];   a.y = G1s[l16][kk + 1];
      b.x = valid ? W1s[n][kk]     : 0.f;  // W1T[kk][n] = W1[n][kk]
      b.y = valid ? W1s[n][kk + 1] : 0.f;
      acc = __builtin_amdgcn_wmma_f32_16x16x4_f32(false, a, false, b,
                                                  (short)0, acc, false, false);
    }
    if (valid) {
#pragma unroll
      for (int v = 0; v < 8; ++v)
        dE_out[(size_t)(atom0 + v + 8 * half) * F_ + n] = acc[v];
    }
  }
}

// ---------------------------------------------------------------------------
// Kernel 2: Etot[b] = sum_n Ei[b,n]  (fixed-order tree => deterministic)
// ---------------------------------------------------------------------------
__global__ __launch_bounds__(256) void mlff_etot_kernel(
    const float* __restrict__ Ei, float* __restrict__ Etot)
{
  const int b = blockIdx.x, tid = threadIdx.x;
  float s = 0.f;
  for (int i = tid; i < N_; i += 256) s += Ei[b * N_ + i];
  __shared__ float red[256];
  red[tid] = s;
  __syncthreads();
  for (int st = 128; st > 0; st >>= 1) {
    if (tid < st) red[tid] += red[tid + st];
    __syncthreads();
  }
  if (tid == 0) Etot[b] = red[0];
}

// ---------------------------------------------------------------------------
// Kernel 3: Force[b,n,:] = sum_{k,f} dE_pad[b, nb(k)][f] * dfeat[b,n,k,f,:]
// One block per atom; streams 50.4 KB of dfeat coalesced; gather rows are
// L2-resident (dE table is 688 KB total). Fixed-order reduction.
// ---------------------------------------------------------------------------
__global__ __launch_bounds__(128) void mlff_force_kernel(
    const float* __restrict__ dE,       // [B*N, F]
    const float* __restrict__ dfeat,    // [B, N, K, F, 3]
    const int*   __restrict__ neighbor, // [B, N, K] 1-based, 0 = none
    float* __restrict__ force)          // [B*N, 3]
{
  const int atom = blockIdx.x;          // flat b*N + n
  const int b    = atom >> 10;          // N_ == 1024
  const int tid  = threadIdx.x;

  __shared__ int nbs[K_];
  if (tid < K_) nbs[tid] = neighbor[(size_t)atom * K_ + tid];
  __syncthreads();

  const float* df = dfeat + (size_t)atom * K_ * F_ * 3;
  float fx = 0.f, fy = 0.f, fz = 0.f;
  for (int idx = tid; idx < K_ * F_; idx += 128) {
    const int k = idx / F_;
    const int f = idx - k * F_;
    const int nbi = nbs[k];
    float w = 0.f;
    if (nbi != 0)
      w = dE[((size_t)b * N_ + (nbi - 1)) * F_ + f];
    const float* d3 = df + (size_t)idx * 3;
    fx += w * d3[0];
    fy += w * d3[1];
    fz += w * d3[2];
  }

  __shared__ float red[3][128];
  red[0][tid] = fx; red[1][tid] = fy; red[2][tid] = fz;
  __syncthreads();
  for (int st = 64; st > 0; st >>= 1) {
    if (tid < st) {
      red[0][tid] += red[0][tid + st];
      red[1][tid] += red[1][tid + st];
      red[2][tid] += red[2][tid + st];
    }
    __syncthreads();
  }
  if (tid == 0) {
    force[(size_t)atom * 3 + 0] = red[0][0];
    force[(size_t)atom * 3 + 1] = red[1][0];
    force[(size_t)atom * 3 + 2] = red[2][0];
  }
}

// ---------------------------------------------------------------------------
// Launch: inputs in setup_inputs() order; d_out = [Etot(4) | Ei(4096) | Force(12288)]
// d_ws holds dE [B*N*F] floats (688 KB).
// ---------------------------------------------------------------------------
extern "C" void kernel_launch(void* const* d_in, const int* in_sizes, int n_in,
                              void* d_out, int out_size, void* d_ws, size_t ws_size,
                              hipStream_t stream) {
  const float* image    = (const float*)d_in[0];
  const float* dfeat    = (const float*)d_in[1];
  const int*   neighbor = (const int*)  d_in[2];
  // d_in[3] Egroup_weight, d_in[4] divider: unused by the reference outputs
  const float* W1 = (const float*)d_in[5];
  const float* b1 = (const float*)d_in[6];
  const float* W2 = (const float*)d_in[7];
  const float* b2 = (const float*)d_in[8];
  const float* W3 = (const float*)d_in[9];
  const float* b3 = (const float*)d_in[10];

  float* out   = (float*)d_out;
  float* Etot  = out;                    // [B]
  float* Ei    = out + B_;               // [B*N]
  float* Force = out + B_ + B_ * N_;     // [B*N*3]
  float* dE    = (float*)d_ws;           // [B*N*F]

  mlff_mlp_wmma_kernel<<<(B_ * N_) / 16, 128, 0, stream>>>(
      image, W1, b1, W2, b2, W3, b3, Ei, dE);
  mlff_etot_kernel<<<B_, 256, 0, stream>>>(Ei, Etot);
  mlff_force_kernel<<<B_ * N_, 128, 0, stream>>>(dE, dfeat, neighbor, Force);
}